// QAttention_17849884082768
// MI455X (gfx1250) — compile-verified
//
#include <hip/hip_runtime.h>

typedef float v2f __attribute__((ext_vector_type(2)));
typedef float v8f __attribute__((ext_vector_type(8)));

// D = A(16x4) * B(4x16) + C, f32 everywhere (args: neg_a, A, neg_b, B, c_mod, C, reuse_a, reuse_b)
#define WMMA_F32(a, b, c) \
  __builtin_amdgcn_wmma_f32_16x16x4_f32(false, (a), false, (b), (short)0, (c), false, false)

__launch_bounds__(256, 1)
__global__ void qattn_fused_kernel(const float* __restrict__ mea_r,
                                   const float* __restrict__ mea_i,
                                   const float* __restrict__ mw,
                                   const float* __restrict__ s_r,
                                   const float* __restrict__ s_i,
                                   float* __restrict__ out)
{
    constexpr int T = 256, D = 64, STR = 66;   // stride-66 pad: conflict-free & 8B aligned
    __shared__ float ST[2][D * STR];           // ST[0]=S_r^T, ST[1]=S_i^T : [e*STR + d]
    __shared__ float Xs[2][32 * STR];          // Xs[0]=Xr rows, Xs[1]=Xi rows (staged)
    __shared__ float prob[T];                  // prob[t], later softmax weights w[t]
    __shared__ float red[T];                   // reduction scratch

    const int tid = threadIdx.x;
    // Broadcast wave id to an SGPR so all per-wave constants below are uniform.
    const int wv  = __builtin_amdgcn_readfirstlane(tid >> 5);   // wave id 0..7
    const int g   = (tid >> 4) & 1;    // lane-half (K split for A/B fragments)
    const int ln  = tid & 15;          // lane within half (M for A, N for B/C/D)
    const int nb  = blockIdx.x;        // n*32 + b
    const int b   = nb & 31;

    const float* xr_g = mea_r + (size_t)b * T * D;
    const float* xi_g = mea_i + (size_t)b * T * D;
    const float* sr_g = s_r + (size_t)nb * D * D;
    const float* si_g = s_i + (size_t)nb * D * D;

    // ---- stage S^T (contiguous-in-d rows -> b64 B-fragments), init prob ----
    for (int i = tid; i < D * D; i += 256) {
        int d = i >> 6, e = i & 63;
        ST[0][e * STR + d] = sr_g[i];
        ST[1][e * STR + d] = si_g[i];
    }
    prob[tid] = 0.0f;
    __syncthreads();

    // ================= Phase 1: prob[t] =================
    // P = Xr*Sr - Xi*Si ; Q = Xr*Si + Xi*Sr ; prob[t] = sum_e P.*Xr + Q.*Xi
    // Waves 0-3: P tiles (e0 = 16*wv). Waves 4-7: Q tiles (e0 = 16*(wv-4)).
    // Branch-free: uniform selects of LDS operands + sign folded into A-fragment,
    // so EXEC stays all-1s through every WMMA.
    const int   isP  = (wv < 4) ? 1 : 0;
    const int   e0p  = (wv & 3) * 16;
    const float sgn  = isP ? -1.0f : 1.0f;          // P: -Xi*Si ; Q: +Xi*Sr
    const float* B1  = ST[1 - isP];                 // P: Sr^T  ; Q: Si^T  (pairs with Ar)
    const float* B2  = ST[isP];                     // P: Si^T  ; Q: Sr^T  (pairs with sgn*Ai)
    const float* Xel = Xs[1 - isP];                 // P: Xr    ; Q: Xi    (elementwise mult)

    for (int t0 = 0; t0 < T; t0 += 16) {
        for (int i = tid; i < 16 * D; i += 256) {     // stage 16 rows of Xr/Xi
            int r = i >> 6, c = i & 63;
            Xs[0][r * STR + c] = xr_g[(t0 + r) * D + c];
            Xs[1][r * STR + c] = xi_g[(t0 + r) * D + c];
        }
        __syncthreads();

        v8f acc = {};
        for (int d0 = 0; d0 < D; d0 += 4) {
            const int dk = d0 + 2 * g;
            v2f ar = *(const v2f*)&Xs[0][ln * STR + dk];          // A[m=ln, k]
            v2f ai = *(const v2f*)&Xs[1][ln * STR + dk];
            ai.x *= sgn;  ai.y *= sgn;
            v2f b1 = *(const v2f*)&B1[(e0p + ln) * STR + dk];     // B[k, n=ln]
            v2f b2 = *(const v2f*)&B2[(e0p + ln) * STR + dk];
            acc = WMMA_F32(ar, b1, acc);
            acc = WMMA_F32(ai, b2, acc);
        }
        // elementwise (.* X) then reduce over the 16 column lanes -> prob[t]
        for (int k = 0; k < 8; ++k) {
            const int r = k + 8 * g;                  // C/D layout: row = k + 8*(lane/16)
            float pv = acc[k] * Xel[r * STR + e0p + ln];
            pv += __shfl_xor(pv, 1, 16);
            pv += __shfl_xor(pv, 2, 16);
            pv += __shfl_xor(pv, 4, 16);
            pv += __shfl_xor(pv, 8, 16);
            if (ln == 0) atomicAdd(&prob[t0 + r], pv);   // ds_add_f32
        }
        __syncthreads();
    }

    // ================= softmax over t (scaled by mea_weights) =================
    float v = prob[tid] * mw[b * T + tid];
    red[tid] = v;
    __syncthreads();
    for (int s = 128; s > 0; s >>= 1) {
        if (tid < s) red[tid] = fmaxf(red[tid], red[tid + s]);
        __syncthreads();
    }
    float mx = red[0];
    __syncthreads();
    float ex = __expf(v - mx);
    red[tid] = ex;
    __syncthreads();
    for (int s = 128; s > 0; s >>= 1) {
        if (tid < s) red[tid] += red[tid + s];
        __syncthreads();
    }
    float sum = red[0];
    __syncthreads();
    prob[tid] = ex / sum;                 // prob[] now holds w[t]
    __syncthreads();

    // ================= Phase 2: out = (diag(w)Xr)^T Xr + (diag(w)Xi)^T Xi ====
    // 4x4 grid of 16x16 output tiles; wave wv owns tiles {wv, wv+8}.
    v8f acc2[2] = {v8f{}, v8f{}};
    const int d0s[2] = { ((wv)     >> 2) * 16, ((wv + 8) >> 2) * 16 };
    const int e0s[2] = { ((wv)     &  3) * 16, ((wv + 8) &  3) * 16 };

    for (int tb = 0; tb < T; tb += 32) {
        for (int i = tid; i < 32 * D; i += 256) {    // stage 32 rows
            int r = i >> 6, c = i & 63;
            Xs[0][r * STR + c] = xr_g[(tb + r) * D + c];
            Xs[1][r * STR + c] = xi_g[(tb + r) * D + c];
        }
        __syncthreads();
        for (int tt = 0; tt < 2; ++tt) {
            const int d0 = d0s[tt], e0 = e0s[tt];
            v8f acc = acc2[tt];
            for (int kk = 0; kk < 32; kk += 4) {     // K = t, steps of 4
                const int tl = kk + 2 * g;           // local t for this lane-half
                const float w0 = prob[tb + tl];
                const float w1 = prob[tb + tl + 1];
                v2f ar, ai, br, bi;
                ar.x = w0 * Xs[0][tl * STR + d0 + ln];        // A[m=d, k=t] = w_t*X[t,d]
                ar.y = w1 * Xs[0][(tl + 1) * STR + d0 + ln];
                ai.x = w0 * Xs[1][tl * STR + d0 + ln];
                ai.y = w1 * Xs[1][(tl + 1) * STR + d0 + ln];
                br.x = Xs[0][tl * STR + e0 + ln];             // B[k=t, n=e] = X[t,e]
                br.y = Xs[0][(tl + 1) * STR + e0 + ln];
                bi.x = Xs[1][tl * STR + e0 + ln];
                bi.y = Xs[1][(tl + 1) * STR + e0 + ln];
                acc = WMMA_F32(ar, br, acc);
                acc = WMMA_F32(ai, bi, acc);
            }
            acc2[tt] = acc;
        }
        __syncthreads();
    }

    // ---- store out_r and out_i (reference makes them identical) ----
    float* out_r = out;
    float* out_i = out + (size_t)2 * 32 * 64 * 64;
    for (int tt = 0; tt < 2; ++tt) {
        const int d0 = d0s[tt], e0 = e0s[tt];
        for (int k = 0; k < 8; ++k) {
            const int row = d0 + k + 8 * g;
            const size_t o = (size_t)nb * (D * D) + (size_t)row * D + e0 + ln;
            const float val = acc2[tt][k];
            out_r[o] = val;
            out_i[o] = val;
        }
    }
}

extern "C" void kernel_launch(void* const* d_in, const int* in_sizes, int n_in,
                              void* d_out, int out_size, void* d_ws, size_t ws_size,
                              hipStream_t stream) {
    (void)in_sizes; (void)n_in; (void)out_size; (void)d_ws; (void)ws_size;
    const float* mea_r = (const float*)d_in[0];   // [32,256,64]
    const float* mea_i = (const float*)d_in[1];   // [32,256,64]
    const float* mw    = (const float*)d_in[2];   // [32,256,1]
    const float* s_r   = (const float*)d_in[3];   // [2,32,64,64]
    const float* s_i   = (const float*)d_in[4];   // [2,32,64,64]
    float* out = (float*)d_out;                   // out_r ++ out_i, 2*262144 floats

    qattn_fused_kernel<<<dim3(64), dim3(256), 0, stream>>>(mea_r, mea_i, mw, s_r, s_i, out);
}